// Encoder_GNN_u_weighted_81071802679528
// MI455X (gfx1250) — compile-verified
//
#include <hip/hip_runtime.h>

// ---------------------------------------------------------------------------
// GraphConv GNN for MI455X (gfx1250, wave32).
//   agg = scatter_add(x_src[src] * w)   -> bandwidth bound, L2-resident targets
//   out = relu(agg @ W_rel + b + x_dst @ W_root)  -> V_WMMA_F32_16X16X4_F32
// GEMM: branch-free, fully unrolled K loop with an explicit depth-2 register
// pipeline so global loads overlap WMMAs; compile-time N so all W/out
// addresses fold to immediate offsets.
// ---------------------------------------------------------------------------

typedef float v2f __attribute__((ext_vector_type(2)));
typedef float v8f __attribute__((ext_vector_type(8)));

#define K_DIM 128

// ---------------------------- elementwise ----------------------------------

__global__ void k_sigmoid(const float* __restrict__ x, float* __restrict__ y, int n) {
  int i = blockIdx.x * blockDim.x + threadIdx.x;
  if (i < n) y[i] = 1.0f / (1.0f + expf(-x[i]));
}

__global__ void k_zero(float4* __restrict__ p, size_t n4) {
  size_t i = (size_t)blockIdx.x * blockDim.x + threadIdx.x;
  size_t stride = (size_t)gridDim.x * blockDim.x;
  float4 z; z.x = z.y = z.z = z.w = 0.0f;
  for (; i < n4; i += stride) p[i] = z;
}

// ---------------------------- edge scatter ---------------------------------
// One wave per edge; each lane moves 4 floats of the 128-float feature row.
// Targets are dense [N_dst,128] buffers that live in the 192MB L2.

template <bool WEIGHTED>
__global__ void k_scatter(const float* __restrict__ x, const int* __restrict__ src,
                          const int* __restrict__ dst, const float* __restrict__ ew,
                          float* __restrict__ agg, int nE) {
  int e = blockIdx.x * (blockDim.x >> 5) + (threadIdx.x >> 5);
  if (e >= nE) return;
  int lane = threadIdx.x & 31;
  int s = src[e];
  int d = dst[e];
  const float4 v = *(const float4*)(x + (size_t)s * K_DIM + lane * 4);
  float w = WEIGHTED ? ew[e] : 1.0f;
  float* p = agg + (size_t)d * K_DIM + lane * 4;
  unsafeAtomicAdd(p + 0, v.x * w);
  unsafeAtomicAdd(p + 1, v.y * w);
  unsafeAtomicAdd(p + 2, v.z * w);
  unsafeAtomicAdd(p + 3, v.w * w);
}

// ---------------------------- WMMA dual GEMM -------------------------------
// out[M,NCOL] = act( A1[M,128] @ W1[128,NCOL] + bias + A2[M,128] @ W2 )
// Per ISA 7.12.2 (f32):
//   A 16x4 : lane l (row = l&15) holds K = k0 + 2*(l>>4) + {0,1}  (float2)
//   B 4x16 : lane l (col = l&15) holds K = k0 + 2*(l>>4) + {0,1}
//   C 16x16: VGPR i, lane l -> row = i + 8*(l>>4), col = l&15
// Wave w owns column tile w; MT row tiles per wave reuse the B registers.
// Grid covers M exactly -> no bounds checks, EXEC all-ones throughout.
// Depth-2 prefetch pipeline: loads for step s+2 issue before WMMAs of step s.

template <bool HAS_ROOT, bool DO_RELU, int MT, int NCOL>
__global__ void k_gemm(const float* __restrict__ A1, const float* __restrict__ A2,
                       const float* __restrict__ W1, const float* __restrict__ W2,
                       const float* __restrict__ bias, float* __restrict__ out) {
  constexpr int NSTEP = K_DIM / 4;  // 32 k-steps
  constexpr int PIPE = 3;           // prefetch distance 2 -> 3 rotating slots

  const int lane = threadIdx.x & 31;
  const int wave = threadIdx.x >> 5;
  const int hi = lane >> 4;
  const int lm = lane & 15;
  const int col = wave * 16 + lm;
  const int rowBase = blockIdx.x * (16 * MT);

  // fold the lane-dependent 2*hi K offset into the base pointers
  const float* a1p = A1 + (size_t)(rowBase + lm) * K_DIM + 2 * hi;
  const float* a2p = HAS_ROOT ? (A2 + (size_t)(rowBase + lm) * K_DIM + 2 * hi) : nullptr;
  const float* w1p = W1 + (size_t)(2 * hi) * NCOL + col;
  const float* w2p = HAS_ROOT ? (W2 + (size_t)(2 * hi) * NCOL + col) : nullptr;

  v2f a1buf[PIPE][MT];
  v2f a2buf[PIPE][MT];
  v2f b1buf[PIPE];
  v2f b2buf[PIPE];
  v8f acc[MT] = {};

  auto load_step = [&](int s, int slot) {
    const int kk = s * 4;  // K offsets are compile-time after full unroll
    v2f b1;
    b1.x = w1p[(size_t)kk * NCOL];
    b1.y = w1p[(size_t)(kk + 1) * NCOL];
    b1buf[slot] = b1;
    if (HAS_ROOT) {
      v2f b2;
      b2.x = w2p[(size_t)kk * NCOL];
      b2.y = w2p[(size_t)(kk + 1) * NCOL];
      b2buf[slot] = b2;
    }
#pragma unroll
    for (int t = 0; t < MT; ++t)
      a1buf[slot][t] = *(const v2f*)(a1p + (size_t)t * 16 * K_DIM + kk);
    if (HAS_ROOT) {
#pragma unroll
      for (int t = 0; t < MT; ++t)
        a2buf[slot][t] = *(const v2f*)(a2p + (size_t)t * 16 * K_DIM + kk);
    }
  };

  load_step(0, 0);
  load_step(1, 1);

#pragma unroll
  for (int s = 0; s < NSTEP; ++s) {
    const int slot = s % PIPE;
    if (s + 2 < NSTEP) load_step(s + 2, (s + 2) % PIPE);
#pragma unroll
    for (int t = 0; t < MT; ++t)
      acc[t] = __builtin_amdgcn_wmma_f32_16x16x4_f32(
          false, a1buf[slot][t], false, b1buf[slot], (short)0, acc[t], false, false);
    if (HAS_ROOT) {
#pragma unroll
      for (int t = 0; t < MT; ++t)
        acc[t] = __builtin_amdgcn_wmma_f32_16x16x4_f32(
            false, a2buf[slot][t], false, b2buf[slot], (short)0, acc[t], false, false);
    }
  }

  const float bv = bias[col];
  float* outp = out + (size_t)rowBase * NCOL + (size_t)(8 * hi) * NCOL + col;
#pragma unroll
  for (int t = 0; t < MT; ++t) {
#pragma unroll
    for (int i = 0; i < 8; ++i) {
      float v = acc[t][i] + bv;
      if (DO_RELU) v = fmaxf(v, 0.0f);
      outp[(size_t)(t * 16 + i) * NCOL] = v;
    }
  }
}

// host-side GEMM dispatch: MT=5 tiles M exactly for all layers (50000/16 and
// 20000/16 are divisible by 5); guard-free MT=1 fallback for other shapes.
template <bool HAS_ROOT, bool DO_RELU, int NCOL>
static void launch_gemm(const float* A1, const float* A2, const float* W1,
                        const float* W2, const float* bias, float* out,
                        int M, hipStream_t stream) {
  const int tiles = M / 16;
  const int threads = (NCOL / 16) * 32;
  if (tiles % 5 == 0) {
    k_gemm<HAS_ROOT, DO_RELU, 5, NCOL><<<tiles / 5, threads, 0, stream>>>(
        A1, A2, W1, W2, bias, out);
  } else {
    k_gemm<HAS_ROOT, DO_RELU, 1, NCOL><<<tiles, threads, 0, stream>>>(
        A1, A2, W1, W2, bias, out);
  }
}

// ---------------------------- launcher -------------------------------------

extern "C" void kernel_launch(void* const* d_in, const int* in_sizes, int n_in,
                              void* d_out, int out_size, void* d_ws, size_t ws_size,
                              hipStream_t stream) {
  const float* x_meas  = (const float*)d_in[0];
  const float* x_dem   = (const float*)d_in[1];
  const int*   src_m   = (const int*)d_in[2];
  const int*   dst_m   = (const int*)d_in[3];
  const int*   src_b   = (const int*)d_in[4];
  const int*   dst_b   = (const int*)d_in[5];
  const float* edge_w  = (const float*)d_in[6];
  const float* W_rel1  = (const float*)d_in[7];
  const float* b_rel1  = (const float*)d_in[8];
  const float* W_root1 = (const float*)d_in[9];
  const float* W_rel2  = (const float*)d_in[10];
  const float* b_rel2  = (const float*)d_in[11];
  const float* W_root2 = (const float*)d_in[12];
  const float* W_rel3  = (const float*)d_in[13];
  const float* b_rel3  = (const float*)d_in[14];
  const float* W_root3 = (const float*)d_in[15];
  const float* W_lin   = (const float*)d_in[16];
  const float* b_lin   = (const float*)d_in[17];
  float* out = (float*)d_out;

  const int E  = in_sizes[2];
  const int NM = in_sizes[0] / K_DIM;  // 50000
  const int ND = in_sizes[1] / K_DIM;  // 20000

  // workspace layout (floats); user_x2 reuses agg_m (dead after layer-1 GEMM)
  float* ws       = (float*)d_ws;
  float* p_ew     = ws;
  float* p_aggm   = p_ew + (((size_t)E + 3) & ~(size_t)3);
  float* p_movie  = p_aggm + (size_t)NM * K_DIM;
  float* p_aggu   = p_movie + (size_t)NM * K_DIM;
  float* p_userx  = p_aggu + (size_t)ND * K_DIM;
  float* p_userx2 = p_aggm;

  const int edgeBlocks = (E + 7) / 8;  // 8 edges (waves) per 256-thr block

  // edge weights: sigmoid
  k_sigmoid<<<(E + 255) / 256, 256, 0, stream>>>(edge_w, p_ew, E);

  // ---- layer 1: movie_x = relu(agg(x_meas) @ Wrel1 + b + x_meas @ Wroot1)
  k_zero<<<2048, 256, 0, stream>>>((float4*)p_aggm, (size_t)NM * K_DIM / 4);
  k_scatter<false><<<edgeBlocks, 256, 0, stream>>>(x_meas, src_m, dst_m, nullptr, p_aggm, E);
  launch_gemm<true, true, 128>(p_aggm, x_meas, W_rel1, W_root1, b_rel1, p_movie, NM, stream);

  // ---- layer 2: user_x = relu(agg_w(x_meas) @ Wrel2 + b + x_dem @ Wroot2)
  k_zero<<<1024, 256, 0, stream>>>((float4*)p_aggu, (size_t)ND * K_DIM / 4);
  k_scatter<true><<<edgeBlocks, 256, 0, stream>>>(x_meas, src_b, dst_b, p_ew, p_aggu, E);
  launch_gemm<true, true, 128>(p_aggu, x_dem, W_rel2, W_root2, b_rel2, p_userx, ND, stream);

  // ---- layer 3: user_x = relu(agg_w(movie_x) @ Wrel3 + b + user_x @ Wroot3)
  k_zero<<<1024, 256, 0, stream>>>((float4*)p_aggu, (size_t)ND * K_DIM / 4);
  k_scatter<true><<<edgeBlocks, 256, 0, stream>>>(p_movie, src_b, dst_b, p_ew, p_aggu, E);
  launch_gemm<true, true, 128>(p_aggu, p_userx, W_rel3, W_root3, b_rel3, p_userx2, ND, stream);

  // ---- final: out = user_x @ W_lin + b_lin   (NCOL = 64 -> 4 waves per block)
  launch_gemm<false, false, 64>(p_userx2, nullptr, W_lin, nullptr, b_lin, out, ND, stream);
}